// MoERouter_6846177870125
// MI455X (gfx1250) — compile-verified
//
#include <hip/hip_runtime.h>
#include <hip/hip_bf16.h>
#include <math.h>

typedef float v2f __attribute__((ext_vector_type(2)));
typedef float v4f __attribute__((ext_vector_type(4)));
typedef float v8f __attribute__((ext_vector_type(8)));

#define HID 4096
#define NEXP 64
#define TOK_PER_BLK 16
#define ROW_STRIDE 68   // 64 + 4 pad: keeps rows 16B aligned, conflict-free b128 scans

// One block = 128 threads = 4 waves. Block owns 16 tokens; wave w owns expert
// tile [16w, 16w+16). GEMM via V_WMMA_F32_16X16X4_F32, K consumed 8 at a time
// with one B128 load for A and one for B per wave per chunk (K-order permuted:
// lanes 0-15 carry physical k+0..3, lanes 16-31 carry k+4..7 — legal because
// the K-reduction is order independent and A/B use the same lane->k mapping).
__global__ __launch_bounds__(128)
void moe_router_wmma_kernel(const float* __restrict__ X,
                            const float* __restrict__ W,
                            float* __restrict__ probs,
                            float* __restrict__ rmap) {
    __shared__ __align__(16) float lgs[TOK_PER_BLK][ROW_STRIDE];
    __shared__ float s_p1[TOK_PER_BLK], s_p2[TOK_PER_BLK];
    __shared__ int   s_i1[TOK_PER_BLK], s_i2[TOK_PER_BLK];

    const int tid  = threadIdx.x;
    const int wave = tid >> 5;      // 0..3 -> expert tile
    const int lane = tid & 31;
    const int half = lane >> 4;     // 0: lanes 0-15, 1: lanes 16-31
    const int l15  = lane & 15;

    const int tok0 = blockIdx.x * TOK_PER_BLK;

    // Per-lane streaming rows: A from token row, B from expert row.
    const float* aRow = X + (size_t)(tok0 + l15) * HID + half * 4;
    const float* bRow = W + (size_t)(wave * 16 + l15) * HID + half * 4;

    // Two independent accumulators: no D->C serialization between the two
    // WMMAs of each K chunk; summed at the end.
    v8f c0 = {0.f, 0.f, 0.f, 0.f, 0.f, 0.f, 0.f, 0.f};
    v8f c1 = {0.f, 0.f, 0.f, 0.f, 0.f, 0.f, 0.f, 0.f};

    #pragma unroll 4
    for (int kb = 0; kb < HID; kb += 8) {
        v4f a = *(const v4f*)(aRow + kb);   // global_load_b128
        v4f b = *(const v4f*)(bRow + kb);   // global_load_b128
        v2f a0 = {a.x, a.y}, a1 = {a.z, a.w};
        v2f b0 = {b.x, b.y}, b1 = {b.z, b.w};
        c0 = __builtin_amdgcn_wmma_f32_16x16x4_f32(
                 false, a0, false, b0, (short)0, c0, false, false);
        c1 = __builtin_amdgcn_wmma_f32_16x16x4_f32(
                 false, a1, false, b1, (short)0, c1, false, false);
    }

    // C layout: VGPR r, lanes 0-15 -> M=r, lanes 16-31 -> M=8+r, N=lane&15.
    const int mBase = half * 8;
    #pragma unroll
    for (int r = 0; r < 8; ++r) {
        lgs[mBase + r][wave * 16 + l15] = c0[r] + c1[r];
    }
    __syncthreads();

    // Branchless top-2 per token (strict > == lowest-index tie-break, matches
    // lax.top_k), then stable softmax over the two selected logits.
    if (tid < TOK_PER_BLK) {
        const v4f* row4 = (const v4f*)&lgs[tid][0];   // conflict-free ds_load_b128
        float v1 = -3.402823466e38f, v2 = -3.402823466e38f;
        int   i1 = 0, i2 = 0;
        #pragma unroll 4
        for (int c = 0; c < NEXP / 4; ++c) {
            v4f v4 = row4[c];
            #pragma unroll
            for (int j = 0; j < 4; ++j) {
                float v = v4[j];
                int   e = c * 4 + j;
                bool g1 = v > v1;
                bool g2 = v > v2;
                float nv2 = g1 ? v1 : (g2 ? v : v2);
                int   ni2 = g1 ? i1 : (g2 ? e : i2);
                float nv1 = g1 ? v : v1;
                int   ni1 = g1 ? e : i1;
                v1 = nv1; i1 = ni1; v2 = nv2; i2 = ni2;
            }
        }
        float ex  = __expf(v2 - v1);      // <= 1
        float inv = 1.0f / (1.0f + ex);
        s_p1[tid] = inv;        // prob of top-1
        s_p2[tid] = ex * inv;   // prob of top-2
        s_i1[tid] = i1;
        s_i2[tid] = i2;
    }
    __syncthreads();

    // Dense scatter: every element of both output tiles is written (no
    // pre-zeroing needed; deterministic across graph replays).
    #pragma unroll
    for (int i = tid; i < TOK_PER_BLK * NEXP; i += 128) {
        int t = i >> 6;        // token within block
        int e = i & (NEXP - 1);
        int i1 = s_i1[t], i2 = s_i2[t];
        float p = (e == i1) ? s_p1[t] : ((e == i2) ? s_p2[t] : 0.0f);
        float m = (e == i1 || e == i2) ? 1.0f : 0.0f;
        size_t off = (size_t)(tok0 + t) * NEXP + e;
        probs[off] = p;
        rmap[off]  = m;
    }
}

extern "C" void kernel_launch(void* const* d_in, const int* in_sizes, int n_in,
                              void* d_out, int out_size, void* d_ws, size_t ws_size,
                              hipStream_t stream) {
    const float* X = (const float*)d_in[0];   // [2048,4,4096] -> [T=8192, 4096]
    const float* W = (const float*)d_in[1];   // [64, 4096]
    float* out = (float*)d_out;

    const int T = in_sizes[0] / HID;          // 8192 tokens
    float* probs = out;                       // [T, 64] fp32
    float* rmap  = out + (size_t)T * NEXP;    // [T, 64] bool-as-float

    dim3 grid(T / TOK_PER_BLK);               // 512 blocks
    moe_router_wmma_kernel<<<grid, 128, 0, stream>>>(X, W, probs, rmap);
}